// MurraySingleArea_39410619908530
// MI455X (gfx1250) — compile-verified
//
#include <hip/hip_runtime.h>
#include <stdint.h>

// ---------------- tunables ----------------
#define BLOCK 64          // threads per block (2 waves, no barriers needed)
#define CH    8           // time-bins per pipeline chunk
#define NS    3           // async pipeline depth (chunks in flight)
#define LDS_TSTRIDE 18    // floats per thread per stage: 16 data + 2 pad (bank-conflict-free b64)
#define STAGE_FLOATS (BLOCK * LDS_TSTRIDE)

// ---------------- model constants ----------------
#define P_A     270.0f
#define P_B     108.0f
#define P_C     0.154f
#define P_DT    0.001f
#define P_I0    0.334f
#define P_TAU   0.06f
#define P_GAMMA 0.641f

typedef __attribute__((ext_vector_type(4))) float v4f;  // native vector for NT b128 stores
typedef __attribute__((ext_vector_type(2))) float v2f;

#if __has_builtin(__builtin_amdgcn_exp2f)
#define EXP2F(x) __builtin_amdgcn_exp2f(x)
#else
#define EXP2F(x) exp2f(x)
#endif

struct SimConsts {
  float AJ00, AJ01, AJ10, AJ11;   // A * J (state@J folded with f-I gain)
  float C1, G, K, invC, AB;
};

// One substep of the 2-state ODE (dependent chain ~10 ops incl. exp2 + rcp).
__device__ __forceinline__ void substep(float& s0, float& s1, float& r0, float& r1,
                                        float ab0, float ab1, const SimConsts& c)
{
  float act0 = fmaf(c.AJ00, s0, fmaf(c.AJ10, s1, ab0));
  float act1 = fmaf(c.AJ01, s0, fmaf(c.AJ11, s1, ab1));
  float u0 = 1.0f - s0, u1 = 1.0f - s1;     // off critical path
  float q0 = c.G * u0,  q1 = c.G * u1;      // off critical path
  float e0 = EXP2F(act0 * c.K);             // exp(-C*act) via v_exp_f32
  float e1 = EXP2F(act1 * c.K);
  float d0 = 1.0f - e0, d1 = 1.0f - e1;
  r0 = act0 * __builtin_amdgcn_rcpf(d0);
  r1 = act1 * __builtin_amdgcn_rcpf(d1);
  r0 = (act0 == 0.0f) ? c.invC : r0;        // limit act->0
  r1 = (act1 == 0.0f) ? c.invC : r1;
  s0 = fmaf(c.C1, s0, q0 * r0);             // s*(1-dt/TAU) + dt*G*(1-s)*r
  s1 = fmaf(c.C1, s1, q1 * r1);
}

// SPB > 0: compile-time unrolled straight-line path. SPB == 0: dynamic count.
template <int SPB>
__device__ __forceinline__ void do_bin(float& s0, float& s1, float& r0, float& r1,
                                       float ab0, float ab1, int spb_dyn,
                                       const SimConsts& c)
{
  if constexpr (SPB > 0) {
#pragma unroll
    for (int k = 0; k < SPB; ++k) substep(s0, s1, r0, r1, ab0, ab1, c);
  } else {
#pragma unroll 4
    for (int k = 0; k < spb_dyn; ++k) substep(s0, s1, r0, r1, ab0, ab1, c);
  }
}

template <int SPB>
__device__ __forceinline__ void run_sim(
    const float* __restrict__ in, float* __restrict__ smem,
    float s0, float s1, int spb_dyn, const SimConsts& cst,
    float* st_out, float* rt_out, float* fin_out,
    size_t rowoff, int b, int T, int tid)
{
  float r0 = 0.0f, r1 = 0.0f;

  const uint64_t gbase = (uint64_t)(uintptr_t)in;
  const unsigned voff0 = (unsigned)((size_t)b * (size_t)T * 8u); // byte offset of row

  // Issue one chunk (CH bins = CH async b64 loads) into its LDS stage.
  auto issue = [&](int c) {
    const unsigned lds_a =
        (unsigned)(uintptr_t)&smem[(c % NS) * STAGE_FLOATS + tid * LDS_TSTRIDE];
    const unsigned voff = voff0 + (unsigned)(c * CH) * 8u;
    asm volatile(
      "global_load_async_to_lds_b64 %0, %1, %2 offset:0\n\t"
      "global_load_async_to_lds_b64 %0, %1, %2 offset:8\n\t"
      "global_load_async_to_lds_b64 %0, %1, %2 offset:16\n\t"
      "global_load_async_to_lds_b64 %0, %1, %2 offset:24\n\t"
      "global_load_async_to_lds_b64 %0, %1, %2 offset:32\n\t"
      "global_load_async_to_lds_b64 %0, %1, %2 offset:40\n\t"
      "global_load_async_to_lds_b64 %0, %1, %2 offset:48\n\t"
      "global_load_async_to_lds_b64 %0, %1, %2 offset:56"
      :: "v"(lds_a), "v"(voff), "s"(gbase) : "memory");
  };

  const int NC  = T / CH;
  const int PRE = (NC < NS) ? NC : NS;
  for (int c = 0; c < PRE; ++c) issue(c);

  for (int c = 0; c < NC; ++c) {
    // Ensure chunk c landed: async loads complete in order; at most
    // min(NS-1, NC-1-c) chunks (8 ops each) were issued after it.
    if (c + 2 < NC)      { asm volatile("s_wait_asynccnt 16" ::: "memory"); }
    else if (c + 1 < NC) { asm volatile("s_wait_asynccnt 8"  ::: "memory"); }
    else                 { asm volatile("s_wait_asynccnt 0"  ::: "memory"); }

    const float* sl = &smem[(c % NS) * STAGE_FLOATS + tid * LDS_TSTRIDE];
    float ss[CH * 2];
    float rr[CH * 2];
#pragma unroll
    for (int j = 0; j < CH; ++j) {
      v2f ip = *(const v2f*)&sl[2 * j];                   // ds_load_b64, conflict-free
      float ab0 = fmaf(P_A, ip.x, cst.AB);
      float ab1 = fmaf(P_A, ip.y, cst.AB);
      do_bin<SPB>(s0, s1, r0, r1, ab0, ab1, spb_dyn, cst);
      ss[2 * j] = s0; ss[2 * j + 1] = s1;
      rr[2 * j] = r0; rr[2 * j + 1] = r1;
    }

    // Drain this stage's LDS reads, then refill it with a future chunk.
    asm volatile("s_wait_dscnt 0" ::: "memory");
    if (c + NS < NC) issue(c + NS);

    // Stream out chunk results (write-once data: nontemporal b128 stores).
    float* sp = st_out + (size_t)(c * CH) * 2;
    float* rp = rt_out + (size_t)(c * CH) * 2;
#pragma unroll
    for (int q = 0; q < CH / 2; ++q) {
      v4f vs = { ss[4*q], ss[4*q+1], ss[4*q+2], ss[4*q+3] };
      v4f vr = { rr[4*q], rr[4*q+1], rr[4*q+2], rr[4*q+3] };
      __builtin_nontemporal_store(vs, (v4f*)sp + q);
      __builtin_nontemporal_store(vr, (v4f*)rp + q);
    }
  }

  // Tail bins if T % CH != 0 (not hit for T=2000): plain loads.
  for (int t = NC * CH; t < T; ++t) {
    float i0 = in[rowoff + 2 * t], i1 = in[rowoff + 2 * t + 1];
    float ab0 = fmaf(P_A, i0, cst.AB);
    float ab1 = fmaf(P_A, i1, cst.AB);
    do_bin<SPB>(s0, s1, r0, r1, ab0, ab1, spb_dyn, cst);
    st_out[2 * t] = s0; st_out[2 * t + 1] = s1;
    rt_out[2 * t] = r0; rt_out[2 * t + 1] = r1;
  }

  fin_out[(size_t)b * 2 + 0] = s0;
  fin_out[(size_t)b * 2 + 1] = s1;
}

__global__ __launch_bounds__(BLOCK) void murray_kernel(
    const float* __restrict__ in, const float* __restrict__ h0,
    const float* __restrict__ Jp, const int* __restrict__ spbp,
    float* __restrict__ out, int B, int T)
{
  __shared__ float smem[NS * STAGE_FLOATS];

  const int tid = threadIdx.x;
  const int b   = blockIdx.x * BLOCK + tid;
  if (b >= B) return;

  const int   spb    = spbp[0];
  const float sub_dt = P_DT / (float)spb;

  SimConsts cst;
  cst.C1   = 1.0f - sub_dt / P_TAU;
  cst.G    = sub_dt * P_GAMMA;
  cst.AJ00 = P_A * Jp[0]; cst.AJ01 = P_A * Jp[1];
  cst.AJ10 = P_A * Jp[2]; cst.AJ11 = P_A * Jp[3];
  cst.AB   = P_A * P_I0 - P_B;                    // per-bin bias constant
  cst.K    = -P_C * 1.44269504088896340736f;      // -C * log2(e)
  cst.invC = 1.0f / P_C;

  const float s0 = h0[(size_t)b * 2 + 0];
  const float s1 = h0[(size_t)b * 2 + 1];

  const size_t rowoff = (size_t)b * (size_t)T * 2;           // float offset of row
  float* st_out  = out + rowoff;                             // states [B,T,2]
  float* rt_out  = out + (size_t)B * (size_t)T * 2 + rowoff; // rates  [B,T,2]
  float* fin_out = out + (size_t)B * (size_t)T * 4;          // final  [1,B,2]

  // Uniform branch: straight-line fully-unrolled path for the reference's
  // steps_per_bin == 4; generic dynamic path otherwise.
  if (spb == 4) {
    run_sim<4>(in, smem, s0, s1, spb, cst, st_out, rt_out, fin_out, rowoff, b, T, tid);
  } else {
    run_sim<0>(in, smem, s0, s1, spb, cst, st_out, rt_out, fin_out, rowoff, b, T, tid);
  }
}

extern "C" void kernel_launch(void* const* d_in, const int* in_sizes, int n_in,
                              void* d_out, int out_size, void* d_ws, size_t ws_size,
                              hipStream_t stream) {
  const float* in  = (const float*)d_in[0];   // [B, T, 2] f32
  const float* h0  = (const float*)d_in[1];   // [1, B, 2] f32
  const float* Jp  = (const float*)d_in[2];   // [2, 2]    f32
  const int*   spb = (const int*)d_in[3];     // scalar int
  float* out = (float*)d_out;                 // states | rates | final, concat

  const int Bsz = in_sizes[1] / 2;
  const int T   = in_sizes[0] / (Bsz * 2);
  const int blocks = (Bsz + BLOCK - 1) / BLOCK;

  murray_kernel<<<blocks, BLOCK, 0, stream>>>(in, h0, Jp, spb, out, Bsz, T);
}